// RNNDecoderWithLayerNorm_49143015801120
// MI455X (gfx1250) — compile-verified
//
#include <hip/hip_runtime.h>
#include <hip/hip_bf16.h>

// Fused RNN decoder (gfx1250 / CDNA5, wave32).
// Exploits h0 == 0 for every layer: the Wh GEMM vanishes (gh == bh).
// All GEMMs run on V_WMMA_F32_16X16X4_F32; activations stay in LDS.

typedef __attribute__((ext_vector_type(2))) float v2f;
typedef __attribute__((ext_vector_type(8))) float v8f;

constexpr int Bsz = 65536, Z = 512, H = 512, F = 64, L = 3;
constexpr int BM    = 64;        // rows of B per workgroup
constexpr int LDST  = H + 4;     // padded LDS row stride (dwords) -> bank-conflict free
constexpr int NWAVE = 8;         // 256 threads / wave32
constexpr float EPS = 1e-5f;

__device__ __forceinline__ float sigmoidf_(float x) {
  return 1.0f / (1.0f + __expf(-x));
}

// D = A(16x4) x B(4x16) + C, fp32 WMMA. A/B = 2 VGPRs, C/D = 8 VGPRs.
__device__ __forceinline__ v8f wmma4(v2f a, v2f b, v8f c) {
  return __builtin_amdgcn_wmma_f32_16x16x4_f32(
      /*neg_a=*/false, a, /*neg_b=*/false, b,
      /*c_mod=*/(short)0, c, /*reuse_a=*/false, /*reuse_b=*/false);
}

// B-operand tile (4x16, KxN) of W^T, i.e. B[k][n] = W[(n0+n)*512 + k].
// Layout: VGPR0/1 hold K={k0,k0+1} in lanes 0-15 and K={k0+2,k0+3} in lanes 16-31,
// so each lane does one 8-byte global load of 2 consecutive K values.
__device__ __forceinline__ v2f load_w_tile(const float* __restrict__ Wrow0,
                                           int k0, int lane) {
  const float2 v = *reinterpret_cast<const float2*>(
      Wrow0 + (lane & 15) * 512 + k0 + ((lane >> 4) << 1));
  v2f r = {v.x, v.y};
  return r;
}

// A-operand tile (16x4, MxK) from padded LDS: one ds_load_b64 per lane.
__device__ __forceinline__ v2f load_a_tile(const float* __restrict__ src,
                                           int m0, int k0, int lane) {
  const float2 v = *reinterpret_cast<const float2*>(
      src + (m0 + (lane & 15)) * LDST + k0 + ((lane >> 4) << 1));
  v2f r = {v.x, v.y};
  return r;
}

__global__ __launch_bounds__(256, 1)
void rnn_decoder_fused(const float* __restrict__ z,
                       const float* __restrict__ W_in,
                       const float* __restrict__ b_in,
                       const float* __restrict__ Wi,
                       const float* __restrict__ bi,
                       const float* __restrict__ bh,
                       const float* __restrict__ gamma,
                       const float* __restrict__ beta,
                       const float* __restrict__ W_out,
                       const float* __restrict__ b_out,
                       float* __restrict__ out) {
  extern __shared__ float smem[];
  float* bufA = smem;              // BM x LDST
  float* bufB = smem + BM * LDST;  // BM x LDST

  const int tid  = threadIdx.x;
  const int lane = tid & 31;
  const int wave = tid >> 5;
  const int row0 = blockIdx.x * BM;

  // ---- stage z tile (64x512) into bufB, padded rows, float4 coalesced ----
  {
    const float4* zg = reinterpret_cast<const float4*>(z + (size_t)row0 * Z);
    #pragma unroll
    for (int i = 0; i < (BM * Z / 4) / 256; ++i) {
      const int q  = i * 256 + tid;   // float4 index
      const int r  = q >> 7;          // 128 float4 per row
      const int c4 = q & 127;
      *reinterpret_cast<float4*>(bufB + r * LDST + c4 * 4) = zg[q];
    }
  }
  __syncthreads();

  // ---- phase 1: x = relu(z @ W_in^T + b_in) -> bufA ----
  for (int nt = wave; nt < H / 16; nt += NWAVE) {
    const int n0 = nt * 16;
    v8f acc[4] = {};
    const float* wn = W_in + (size_t)n0 * Z;
    for (int k0 = 0; k0 < Z; k0 += 4) {
      const v2f bt = load_w_tile(wn, k0, lane);
      #pragma unroll
      for (int mt = 0; mt < 4; ++mt) {
        const v2f at = load_a_tile(bufB, mt * 16, k0, lane);
        acc[mt] = wmma4(at, bt, acc[mt]);
      }
    }
    const int col = n0 + (lane & 15);
    const float bias = b_in[col];
    #pragma unroll
    for (int mt = 0; mt < 4; ++mt) {
      #pragma unroll
      for (int d = 0; d < 8; ++d) {
        const int row = mt * 16 + d + ((lane >> 4) << 3);
        const float v = acc[mt][d] + bias;
        bufA[row * LDST + col] = v > 0.f ? v : 0.f;
      }
    }
  }
  __syncthreads();

  // ---- 3 GRU(h=0) + residual + LayerNorm layers, ping-ponging LDS buffers ----
  const float* src = bufA;
  float*       dst = bufB;
  for (int l = 0; l < L; ++l) {
    const float* Wl  = Wi + (size_t)l * 3 * H * H;
    const float* bil = bi + l * 3 * H;
    const float* bhl = bh + l * 3 * H;

    for (int nt = wave; nt < H / 16; nt += NWAVE) {
      const int n0 = nt * 16;
      v8f ar[4] = {}, au[4] = {}, an[4] = {};
      const float* wr = Wl + (size_t)(n0) * H;           // reset-gate rows
      const float* wu = Wl + (size_t)(H + n0) * H;       // update-gate rows
      const float* wg = Wl + (size_t)(2 * H + n0) * H;   // new-gate rows
      for (int k0 = 0; k0 < H; k0 += 4) {
        const v2f br = load_w_tile(wr, k0, lane);
        const v2f bu = load_w_tile(wu, k0, lane);
        const v2f bn = load_w_tile(wg, k0, lane);
        #pragma unroll
        for (int mt = 0; mt < 4; ++mt) {
          const v2f at = load_a_tile(src, mt * 16, k0, lane);
          ar[mt] = wmma4(at, br, ar[mt]);
          au[mt] = wmma4(at, bu, au[mt]);
          an[mt] = wmma4(at, bn, an[mt]);
        }
      }
      const int col = n0 + (lane & 15);
      const float bir = bil[col], biu = bil[H + col], big = bil[2 * H + col];
      const float bhr = bhl[col], bhu = bhl[H + col], bhg = bhl[2 * H + col];
      #pragma unroll
      for (int mt = 0; mt < 4; ++mt) {
        #pragma unroll
        for (int d = 0; d < 8; ++d) {
          const int row = mt * 16 + d + ((lane >> 4) << 3);
          const float r  = sigmoidf_(ar[mt][d] + bir + bhr);
          const float u  = sigmoidf_(au[mt][d] + biu + bhu);
          const float nn = tanhf(an[mt][d] + big + r * bhg);
          const float g  = (1.f - u) * nn;            // + u*h0, h0 == 0
          dst[row * LDST + col] = g + src[row * LDST + col];
        }
      }
    }
    __syncthreads();

    // row-wise LayerNorm in-place in dst: 4 lanes per row, shfl_xor reduce
    {
      const int row = tid >> 2, s = tid & 3;
      float* rp = dst + row * LDST + s * (H / 4);
      const float* gl = gamma + l * H + s * (H / 4);
      const float* bl = beta  + l * H + s * (H / 4);
      float sum = 0.f, sq = 0.f;
      #pragma unroll 4
      for (int c = 0; c < H / 4; c += 4) {
        const float4 v = *reinterpret_cast<const float4*>(rp + c);
        sum += v.x + v.y + v.z + v.w;
        sq  += v.x * v.x + v.y * v.y + v.z * v.z + v.w * v.w;
      }
      sum += __shfl_xor(sum, 1); sum += __shfl_xor(sum, 2);
      sq  += __shfl_xor(sq, 1);  sq  += __shfl_xor(sq, 2);
      const float mu  = sum * (1.f / H);
      const float inv = rsqrtf(sq * (1.f / H) - mu * mu + EPS);
      #pragma unroll 4
      for (int c = 0; c < H / 4; c += 4) {
        float4 v = *reinterpret_cast<float4*>(rp + c);
        v.x = (v.x - mu) * inv * gl[c + 0] + bl[c + 0];
        v.y = (v.y - mu) * inv * gl[c + 1] + bl[c + 1];
        v.z = (v.z - mu) * inv * gl[c + 2] + bl[c + 2];
        v.w = (v.w - mu) * inv * gl[c + 3] + bl[c + 3];
        *reinterpret_cast<float4*>(rp + c) = v;
      }
    }
    __syncthreads();

    float* t = const_cast<float*>(src); src = dst; dst = t;
  }

  // ---- out = x @ W_out^T + b_out ; x lives in `src` (bufB after 3 swaps) ----
  {
    const int nt = wave & 3;                 // 4 N-tiles (F=64)
    const int mtBase = (wave >> 2) << 1;     // 2 M-tiles per wave
    const int n0 = nt * 16;
    v8f acc[2] = {};
    const float* wn = W_out + (size_t)n0 * H;
    for (int k0 = 0; k0 < H; k0 += 4) {
      const v2f bt = load_w_tile(wn, k0, lane);
      #pragma unroll
      for (int i = 0; i < 2; ++i) {
        const v2f at = load_a_tile(src, (mtBase + i) * 16, k0, lane);
        acc[i] = wmma4(at, bt, acc[i]);
      }
    }
    const int col = n0 + (lane & 15);
    const float bias = b_out[col];
    #pragma unroll
    for (int i = 0; i < 2; ++i) {
      #pragma unroll
      for (int d = 0; d < 8; ++d) {
        const int row = (mtBase + i) * 16 + d + ((lane >> 4) << 3);
        out[(size_t)(row0 + row) * F + col] = acc[i][d] + bias;
      }
    }
  }
}

extern "C" void kernel_launch(void* const* d_in, const int* in_sizes, int n_in,
                              void* d_out, int out_size, void* d_ws, size_t ws_size,
                              hipStream_t stream) {
  const float* z     = (const float*)d_in[0];
  const float* W_in  = (const float*)d_in[1];
  const float* b_in  = (const float*)d_in[2];
  const float* Wi    = (const float*)d_in[3];
  // d_in[4] = Wh is mathematically unused: h0 == 0 for every layer.
  const float* bi    = (const float*)d_in[5];
  const float* bh    = (const float*)d_in[6];
  const float* gamma = (const float*)d_in[7];
  const float* beta  = (const float*)d_in[8];
  const float* W_out = (const float*)d_in[9];
  const float* b_out = (const float*)d_in[10];
  float* out = (float*)d_out;

  const size_t shmem = (size_t)2 * BM * LDST * sizeof(float);  // 258 KB < 320 KB/WGP
  hipFuncSetAttribute((const void*)rnn_decoder_fused,
                      hipFuncAttributeMaxDynamicSharedMemorySize, (int)shmem);
  rnn_decoder_fused<<<Bsz / BM, 256, shmem, stream>>>(
      z, W_in, b_in, Wi, bi, bh, gamma, beta, W_out, b_out, out);
}